// Autoencoder_34995393527840
// MI455X (gfx1250) — compile-verified
//
#include <hip/hip_runtime.h>
#include <math.h>

// ---------------------------------------------------------------------------
// Problem constants (from reference): N=4096 samples, D=784 features,
// k = N/nnfactor = 64 -> top_k(63), keep 62 neighbors after dropping first.
// ---------------------------------------------------------------------------
#define NPTS 4096
#define DIM  784
#define NNB  62            // neighbors kept (63 top-k minus the first)

typedef float v2f __attribute__((ext_vector_type(2)));
typedef float v8f __attribute__((ext_vector_type(8)));

// ---------------------------------------------------------------------------
// Kernel 1: squared row norms of x  (sq[i] = sum_k x[i,k]^2)
// ---------------------------------------------------------------------------
__global__ __launch_bounds__(256) void sqnorm_kernel(const float* __restrict__ x,
                                                     float* __restrict__ sq) {
    __shared__ float sV[256];
    const int tid = threadIdx.x;
    const int i   = blockIdx.x;
    float s = 0.f;
    for (int k = tid; k < DIM; k += 256) {
        float v = x[(size_t)i * DIM + k];
        s += v * v;
    }
    sV[tid] = s; __syncthreads();
    for (int st = 128; st > 0; st >>= 1) {
        if (tid < st) sV[tid] += sV[tid + st];
        __syncthreads();
    }
    if (tid == 0) sq[i] = sV[0];
}

// ---------------------------------------------------------------------------
// Kernel 2: fused CNN encoder. One block per image; all intermediates in LDS
// (784 + 6272 + 1568 + 784 floats = 37.6KB of the 320KB/WGP LDS).
// conv(3x3, SAME) + relu + maxpool2 stages, then dense 784->16.
// ---------------------------------------------------------------------------
__global__ __launch_bounds__(256) void encoder_kernel(
    const float* __restrict__ x,
    const float* __restrict__ w1, const float* __restrict__ b1,   // (3,3,1,32)
    const float* __restrict__ w2, const float* __restrict__ b2,   // (3,3,32,32)
    const float* __restrict__ w3, const float* __restrict__ b3,   // (3,3,32,16)
    const float* __restrict__ dw, const float* __restrict__ db,   // (784,16)
    float* __restrict__ enc) {
    __shared__ float img[28 * 28];
    __shared__ float h1[14 * 14 * 32];
    __shared__ float h2[7 * 7 * 32];
    __shared__ float h3[7 * 7 * 16];

    const int tid = threadIdx.x;
    const int n   = blockIdx.x;

    for (int i = tid; i < 784; i += 256) img[i] = x[(size_t)n * DIM + i];
    __syncthreads();

    // ---- conv1 (1->32) + relu + pool2 : 28x28 -> 14x14x32 ----
    for (int o = tid; o < 14 * 14 * 32; o += 256) {
        const int oc = o & 31;
        const int px = (o >> 5) % 14;
        const int py = o / (32 * 14);
        float mv = 0.f;                      // relu outputs are >= 0
        for (int dy = 0; dy < 2; ++dy)
            for (int dx = 0; dx < 2; ++dx) {
                const int y = py * 2 + dy, xx = px * 2 + dx;
                float s = b1[oc];
                #pragma unroll
                for (int ky = 0; ky < 3; ++ky) {
                    const int iy = y + ky - 1;
                    if (iy < 0 || iy >= 28) continue;
                    #pragma unroll
                    for (int kx = 0; kx < 3; ++kx) {
                        const int ix = xx + kx - 1;
                        if (ix < 0 || ix >= 28) continue;
                        s += img[iy * 28 + ix] * w1[(ky * 3 + kx) * 32 + oc];
                    }
                }
                mv = fmaxf(mv, fmaxf(s, 0.f));
            }
        h1[(py * 14 + px) * 32 + oc] = mv;
    }
    __syncthreads();

    // ---- conv2 (32->32) + relu + pool2 : 14x14 -> 7x7x32 ----
    for (int o = tid; o < 7 * 7 * 32; o += 256) {
        const int oc = o & 31;
        const int px = (o >> 5) % 7;
        const int py = o / (32 * 7);
        float mv = 0.f;
        for (int dy = 0; dy < 2; ++dy)
            for (int dx = 0; dx < 2; ++dx) {
                const int y = py * 2 + dy, xx = px * 2 + dx;
                float s = b2[oc];
                for (int ky = 0; ky < 3; ++ky) {
                    const int iy = y + ky - 1;
                    if (iy < 0 || iy >= 14) continue;
                    for (int kx = 0; kx < 3; ++kx) {
                        const int ix = xx + kx - 1;
                        if (ix < 0 || ix >= 14) continue;
                        const float* hp = &h1[(iy * 14 + ix) * 32];
                        const float* wp = &w2[((ky * 3 + kx) * 32) * 32 + oc];
                        #pragma unroll
                        for (int ic = 0; ic < 32; ++ic) s += hp[ic] * wp[ic * 32];
                    }
                }
                mv = fmaxf(mv, fmaxf(s, 0.f));
            }
        h2[(py * 7 + px) * 32 + oc] = mv;
    }
    __syncthreads();

    // ---- conv3 (32->16) + relu : 7x7x32 -> 7x7x16 ----
    for (int o = tid; o < 7 * 7 * 16; o += 256) {
        const int oc = o & 15;
        const int px = (o >> 4) % 7;
        const int py = o / (16 * 7);
        float s = b3[oc];
        for (int ky = 0; ky < 3; ++ky) {
            const int iy = py + ky - 1;
            if (iy < 0 || iy >= 7) continue;
            for (int kx = 0; kx < 3; ++kx) {
                const int ix = px + kx - 1;
                if (ix < 0 || ix >= 7) continue;
                const float* hp = &h2[(iy * 7 + ix) * 32];
                const float* wp = &w3[((ky * 3 + kx) * 32) * 16 + oc];
                #pragma unroll
                for (int ic = 0; ic < 32; ++ic) s += hp[ic] * wp[ic * 16];
            }
        }
        h3[(py * 7 + px) * 16 + oc] = fmaxf(s, 0.f);
    }
    __syncthreads();

    // ---- dense 784 -> 16 (no activation) ----
    if (tid < 16) {
        float s = db[tid];
        for (int k = 0; k < 784; ++k) s += h3[k] * dw[k * 16 + tid];
        enc[n * 16 + tid] = s;
    }
}

// ---------------------------------------------------------------------------
// Kernel 3: d_ini = sqrt(max(sq_i + sq_j - 2 * x @ x.T, 1e-12))
// fp32 WMMA (V_WMMA_F32_16X16X4_F32) tiled GEMM: 128x128 block tile,
// 8 waves (2x4), each wave computes 64x32 = 4x2 tiles of 16x16.
// Tile staging uses CDNA5 async global->LDS copies (GLOBAL_LOAD_ASYNC_TO_LDS,
// ASYNCcnt) with double-buffered LDS so the fetch of chunk c+1 overlaps the
// WMMAs of chunk c. LDS stride padded to 20 floats -> conflict-free frag reads.
// ---------------------------------------------------------------------------
__global__ __launch_bounds__(256) void gemm_dist_kernel(
    const float* __restrict__ x, const float* __restrict__ sq,
    float* __restrict__ dout) {
    __shared__ __align__(16) float As[2][128 * 20];
    __shared__ __align__(16) float Bs[2][128 * 20];

    const int tid   = threadIdx.x;
    const int wave  = tid >> 5;
    const int lane  = tid & 31;
    const int waveM = wave & 1;          // 0..1  (rows, 64 each)
    const int waveN = wave >> 1;         // 0..3  (cols, 32 each)
    const int lm    = lane & 15;         // 0..15
    const int kh    = (lane >> 4) * 2;   // 0 or 2 (A/B fragment K offset)
    const int rowBase = blockIdx.y * 128;
    const int colBase = blockIdx.x * 128;

    // per-thread staging coordinates: 512 float4 per matrix / 256 threads
    const int r0 = tid >> 2;             // q = 0
    const int k0 = (tid & 3) * 4;
    const int r1 = (tid + 256) >> 2;     // q = 1
    const int k1 = ((tid + 256) & 3) * 4;

    v8f acc[4][2];
    #pragma unroll
    for (int mi = 0; mi < 4; ++mi)
        #pragma unroll
        for (int ni = 0; ni < 2; ++ni)
            #pragma unroll
            for (int r = 0; r < 8; ++r) acc[mi][ni][r] = 0.f;

    // issue 4 async b128 global->LDS transfers for one 128x16 A/B tile pair
    auto issue_tile = [&](int buf, int kb) {
        {
            const float* ga = x + (size_t)(rowBase + r0) * DIM + kb + k0;
            const unsigned la = (unsigned)(uintptr_t)&As[buf][r0 * 20 + k0];
            asm volatile("global_load_async_to_lds_b128 %0, %1, off"
                         :: "v"(la), "v"(ga) : "memory");
            const float* gb = x + (size_t)(colBase + r0) * DIM + kb + k0;
            const unsigned lb = (unsigned)(uintptr_t)&Bs[buf][r0 * 20 + k0];
            asm volatile("global_load_async_to_lds_b128 %0, %1, off"
                         :: "v"(lb), "v"(gb) : "memory");
        }
        {
            const float* ga = x + (size_t)(rowBase + r1) * DIM + kb + k1;
            const unsigned la = (unsigned)(uintptr_t)&As[buf][r1 * 20 + k1];
            asm volatile("global_load_async_to_lds_b128 %0, %1, off"
                         :: "v"(la), "v"(ga) : "memory");
            const float* gb = x + (size_t)(colBase + r1) * DIM + kb + k1;
            const unsigned lb = (unsigned)(uintptr_t)&Bs[buf][r1 * 20 + k1];
            asm volatile("global_load_async_to_lds_b128 %0, %1, off"
                         :: "v"(lb), "v"(gb) : "memory");
        }
    };

    const int nChunks = DIM / 16;        // 49
    issue_tile(0, 0);

    for (int c = 0; c < nChunks; ++c) {
        const int p = c & 1;
        // wait for this buffer's async transfers, then workgroup handoff
        asm volatile("s_wait_asynccnt 0x0" ::: "memory");
        __syncthreads();
        // prefetch next chunk into the other buffer (overlaps the WMMAs below;
        // safe: buf[1-p] was last read in iteration c-1, fenced by barrier)
        if (c + 1 < nChunks) issue_tile(1 - p, (c + 1) * 16);

        const float* __restrict__ Ap = As[p];
        const float* __restrict__ Bp = Bs[p];
        #pragma unroll
        for (int kk = 0; kk < 16; kk += 4) {
            v2f a[4], b[2];
            #pragma unroll
            for (int mi = 0; mi < 4; ++mi) {
                const int m = waveM * 64 + mi * 16 + lm;
                a[mi].x = Ap[m * 20 + kk + kh];
                a[mi].y = Ap[m * 20 + kk + kh + 1];
            }
            #pragma unroll
            for (int ni = 0; ni < 2; ++ni) {
                const int nn = waveN * 32 + ni * 16 + lm;
                b[ni].x = Bp[nn * 20 + kk + kh];
                b[ni].y = Bp[nn * 20 + kk + kh + 1];
            }
            #pragma unroll
            for (int mi = 0; mi < 4; ++mi)
                #pragma unroll
                for (int ni = 0; ni < 2; ++ni)
                    acc[mi][ni] = __builtin_amdgcn_wmma_f32_16x16x4_f32(
                        false, a[mi], false, b[ni], (short)0, acc[mi][ni],
                        false, false);
        }
        __syncthreads();   // all waves done reading buf[p] before it reloads
    }

    // epilogue: D layout -> lanes 0-15 hold M=r, lanes 16-31 hold M=r+8
    const int rsel = (lane >> 4) * 8;
    #pragma unroll
    for (int mi = 0; mi < 4; ++mi)
        #pragma unroll
        for (int ni = 0; ni < 2; ++ni) {
            const int gn  = colBase + waveN * 32 + ni * 16 + lm;
            const float sqn = sq[gn];
            #pragma unroll
            for (int r = 0; r < 8; ++r) {
                const int gm = rowBase + waveM * 64 + mi * 16 + r + rsel;
                const float d2 = sq[gm] + sqn - 2.f * acc[mi][ni][r];
                dout[(size_t)gm * NPTS + gn] = sqrtf(fmaxf(d2, 1e-12f));
            }
        }
}

// ---------------------------------------------------------------------------
// Kernel 4: per-row top-63 (drop the largest -> 62 neighbors), then
// encoded-space distances for those neighbors and per-row ratio partial sum.
// One block per row; row cached in LDS; deterministic tree reductions with
// lowest-index tie-breaking (matches jax.lax.top_k).
// ---------------------------------------------------------------------------
__global__ __launch_bounds__(256) void topk_kernel(
    const float* __restrict__ din, const float* __restrict__ enc,
    float* __restrict__ valsIni, int* __restrict__ idxOut,
    float* __restrict__ valsEnc, float* __restrict__ ratioPart) {
    __shared__ float row[NPTS];
    __shared__ float sV[256];
    __shared__ int   sI[256];
    __shared__ float selV[NNB];
    __shared__ int   selI[NNB];
    __shared__ float encI[16];

    const int tid = threadIdx.x;
    const int i   = blockIdx.x;

    for (int j = tid; j < NPTS; j += 256) row[j] = din[(size_t)i * NPTS + j];
    if (tid < 16) encI[tid] = enc[i * 16 + tid];
    __syncthreads();

    const int base = tid * 16;
    for (int t = 0; t < NNB + 1; ++t) {      // 63 selections
        float bv = -1.f; int bi = base;
        #pragma unroll
        for (int e = 0; e < 16; ++e) {
            const float v = row[base + e];
            if (v > bv) { bv = v; bi = base + e; }   // keeps lowest index on tie
        }
        sV[tid] = bv; sI[tid] = bi; __syncthreads();
        for (int s = 128; s > 0; s >>= 1) {
            if (tid < s) {
                const float v2 = sV[tid + s]; const int i2 = sI[tid + s];
                if (v2 > sV[tid] || (v2 == sV[tid] && i2 < sI[tid])) {
                    sV[tid] = v2; sI[tid] = i2;
                }
            }
            __syncthreads();
        }
        const int   win = sI[0];
        const float wv  = sV[0];
        if (tid == 0 && t > 0) { selV[t - 1] = wv; selI[t - 1] = win; }
        if (tid == (win >> 4)) row[win] = -1.f;     // remove from candidates
        __syncthreads();
    }

    float ratio = 0.f;
    if (tid < NNB) {
        const int   j  = selI[tid];
        const float vi = selV[tid];
        float ss = 0.f;
        #pragma unroll
        for (int c = 0; c < 16; ++c) {
            const float d = encI[c] - enc[j * 16 + c];
            ss += d * d;
        }
        const float ve = sqrtf(fmaxf(ss, 1e-12f));
        valsIni[(size_t)i * NNB + tid] = vi;
        idxOut [(size_t)i * NNB + tid] = j;
        valsEnc[(size_t)i * NNB + tid] = ve;
        ratio = vi / ve;
    }
    sV[tid] = ratio; __syncthreads();
    for (int s = 128; s > 0; s >>= 1) {
        if (tid < s) sV[tid] += sV[tid + s];
        __syncthreads();
    }
    if (tid == 0) ratioPart[i] = sV[0];
}

// ---------------------------------------------------------------------------
// Kernel 5: multiple = mean(ratio); loss = sum_i max_j (vi - ve*multiple)^2 / N
// Single block, fixed-order reductions (deterministic, no float atomics).
// ---------------------------------------------------------------------------
__global__ __launch_bounds__(256) void finalize_kernel(
    const float* __restrict__ valsIni, const float* __restrict__ valsEnc,
    const float* __restrict__ ratioPart, float* __restrict__ out) {
    __shared__ float sV[256];
    __shared__ float sMult;
    const int tid = threadIdx.x;

    float s = 0.f;
    for (int r = tid; r < NPTS; r += 256) s += ratioPart[r];
    sV[tid] = s; __syncthreads();
    for (int st = 128; st > 0; st >>= 1) {
        if (tid < st) sV[tid] += sV[tid + st];
        __syncthreads();
    }
    if (tid == 0) sMult = sV[0] / ((float)NPTS * (float)NNB);
    __syncthreads();
    const float mult = sMult;

    float tot = 0.f;
    for (int r = tid; r < NPTS; r += 256) {
        float mv = 0.f;                       // squares are >= 0
        for (int j = 0; j < NNB; ++j) {
            const float d = valsIni[(size_t)r * NNB + j]
                          - valsEnc[(size_t)r * NNB + j] * mult;
            mv = fmaxf(mv, d * d);
        }
        tot += mv;
    }
    sV[tid] = tot; __syncthreads();
    for (int st = 128; st > 0; st >>= 1) {
        if (tid < st) sV[tid] += sV[tid + st];
        __syncthreads();
    }
    if (tid == 0) out[0] = sV[0] / (float)NPTS;
}

// ---------------------------------------------------------------------------
// Launch. Inputs (setup_inputs order): x, cw1, cb1, cw2, cb2, cw3, cb3,
// dw, db, nnfactor. Output: scalar f32. Workspace layout (~67.2 MB):
//   d_ini[N*N] | sq[N] | enc[N*16] | vals_ini[N*62] | idx[N*62] |
//   vals_enc[N*62] | ratio_part[N]
// ---------------------------------------------------------------------------
extern "C" void kernel_launch(void* const* d_in, const int* in_sizes, int n_in,
                              void* d_out, int out_size, void* d_ws, size_t ws_size,
                              hipStream_t stream) {
    const float* x   = (const float*)d_in[0];
    const float* cw1 = (const float*)d_in[1];
    const float* cb1 = (const float*)d_in[2];
    const float* cw2 = (const float*)d_in[3];
    const float* cb2 = (const float*)d_in[4];
    const float* cw3 = (const float*)d_in[5];
    const float* cb3 = (const float*)d_in[6];
    const float* dw  = (const float*)d_in[7];
    const float* db  = (const float*)d_in[8];

    float* ws = (float*)d_ws;
    float* dini      = ws;                                  // N*N
    float* sq        = dini + (size_t)NPTS * NPTS;          // N
    float* enc       = sq + NPTS;                           // N*16
    float* valsIni   = enc + (size_t)NPTS * 16;             // N*62
    int*   idxv      = (int*)(valsIni + (size_t)NPTS * NNB);// N*62
    float* valsEnc   = (float*)(idxv + (size_t)NPTS * NNB); // N*62
    float* ratioPart = valsEnc + (size_t)NPTS * NNB;        // N

    sqnorm_kernel<<<NPTS, 256, 0, stream>>>(x, sq);
    encoder_kernel<<<NPTS, 256, 0, stream>>>(x, cw1, cb1, cw2, cb2, cw3, cb3,
                                             dw, db, enc);
    gemm_dist_kernel<<<dim3(NPTS / 128, NPTS / 128), 256, 0, stream>>>(x, sq, dini);
    topk_kernel<<<NPTS, 256, 0, stream>>>(dini, enc, valsIni, idxv, valsEnc,
                                          ratioPart);
    finalize_kernel<<<1, 256, 0, stream>>>(valsIni, valsEnc, ratioPart,
                                           (float*)d_out);
}